// SelfAttention_29987461661287
// MI455X (gfx1250) — compile-verified
//
#include <hip/hip_runtime.h>
#include <math.h>

#define DIM    4096
#define NH     32
#define NKV    8
#define NREP   4
#define HD     128
#define BLOCKK 256
#define TOPK   16
#define NCHUNK (TOPK + 1)        // 16 cached chunks + 1 new-token "chunk"
#define SCALE  0.08838834764831845f   // 1/sqrt(128)

typedef float v2f __attribute__((ext_vector_type(2)));
typedef float v8f __attribute__((ext_vector_type(8)));

// ---------------------------------------------------------------------------
// GEMV: y[r] = dot(W[r, 0:4096], x[0:4096]); one wave32 per row.
// Memory-bound: coalesced float4 (global_load_b128) + prefetch.
// ---------------------------------------------------------------------------
__global__ void gemv4096_kernel(const float* __restrict__ W,
                                const float* __restrict__ x,
                                float* __restrict__ y, int R) {
  const int wave = threadIdx.x >> 5;
  const int lane = threadIdx.x & 31;
  const int row  = blockIdx.x * 8 + wave;
  if (row >= R) return;
  const float4* __restrict__ w4 = (const float4*)(W + (size_t)row * DIM);
  const float4* __restrict__ x4 = (const float4*)x;
  float acc = 0.f;
#pragma unroll 4
  for (int i = 0; i < DIM / 128; ++i) {
    __builtin_prefetch(&w4[(i + 8) * 32 + lane], 0, 0);   // global_prefetch_b8
    float4 wv = w4[i * 32 + lane];
    float4 xv = x4[i * 32 + lane];
    acc = fmaf(wv.x, xv.x, acc);
    acc = fmaf(wv.y, xv.y, acc);
    acc = fmaf(wv.z, xv.z, acc);
    acc = fmaf(wv.w, xv.w, acc);
  }
#pragma unroll
  for (int off = 16; off > 0; off >>= 1)
    acc += __shfl_xor(acc, off, 32);
  if (lane == 0) y[row] = acc;
}

// ---------------------------------------------------------------------------
// RoPE in place on q[32][128] and k[8][128] (interleaved even/odd pairs).
// ---------------------------------------------------------------------------
__global__ void rope_kernel(float* __restrict__ q, float* __restrict__ k,
                            const float* __restrict__ cos_t,
                            const float* __restrict__ sin_t) {
  int tid = blockIdx.x * blockDim.x + threadIdx.x;   // 2560 pairs total
  float* p;
  int d;
  if (tid < NH * (HD / 2)) {
    int h = tid >> 6; d = tid & 63;
    p = q + h * HD + d * 2;
  } else if (tid < (NH + NKV) * (HD / 2)) {
    int t = tid - NH * (HD / 2);
    int h = t >> 6; d = t & 63;
    p = k + h * HD + d * 2;
  } else {
    return;
  }
  float c = cos_t[d], s = sin_t[d];
  float e = p[0], o = p[1];
  p[0] = e * c - o * s;
  p[1] = e * s + o * c;
}

// ---------------------------------------------------------------------------
// Flash-decode partials. Grid: NKV * NCHUNK blocks, 128 threads (4 waves).
// Chunk < TOPK : 256 cached keys from block_idx[chunk].
// Chunk == TOPK: the single appended new token.
// Scores: V_WMMA_F32_16X16X4_F32, A = K_tile(16x4-slice), B = Q^T staged in
// LDS padded to all 16 columns (cols 4..15 zero) so the B fragment is a
// single unconditional ds_load_b64 — no EXEC masking in the WMMA loop.
// ---------------------------------------------------------------------------
__global__ void attn_partial_kernel(const float* __restrict__ q,      // [32][128] roped
                                    const float* __restrict__ k_new,  // [8][128] roped
                                    const float* __restrict__ v_new,  // [8][128]
                                    const float* __restrict__ k_cache,// [NBLK][256][8][128]
                                    const float* __restrict__ v_cache,
                                    const int* __restrict__ block_idx,
                                    float* __restrict__ part_m,       // [32][NCHUNK]
                                    float* __restrict__ part_l,       // [32][NCHUNK]
                                    float* __restrict__ part_acc)     // [32][NCHUNK][128]
{
  const int kvh   = blockIdx.x / NCHUNK;   // 0..7
  const int chunk = blockIdx.x % NCHUNK;   // 0..16
  const int wave  = threadIdx.x >> 5;      // 0..3
  const int lane  = threadIdx.x & 31;

  __shared__ float q_s[16][HD];        // 16 WMMA B-columns: 4 q heads + 12 zero
  __shared__ float s_s[NREP][BLOCKK];  // scores -> exp weights

  // stage the 4 query heads into LDS columns 0..3; zero-fill columns 4..15
  for (int idx = threadIdx.x; idx < 16 * HD; idx += blockDim.x) {
    const int col = idx / HD, d = idx % HD;
    q_s[col][d] = (col < NREP) ? q[(kvh * NREP + col) * HD + d] : 0.f;
  }
  __syncthreads();

  if (chunk == TOPK) {
    // ---- single new-token key: wave == query head ----
    const int qh = wave;
    float s = 0.f;
#pragma unroll
    for (int j = 0; j < 4; ++j)
      s = fmaf(q_s[qh][lane * 4 + j], k_new[kvh * HD + lane * 4 + j], s);
#pragma unroll
    for (int off = 16; off > 0; off >>= 1)
      s += __shfl_xor(s, off, 32);
    s *= SCALE;
    const int gh = kvh * NREP + qh;
    if (lane == 0) {
      part_m[gh * NCHUNK + chunk] = s;     // m = s, p = exp(0) = 1
      part_l[gh * NCHUNK + chunk] = 1.f;
    }
    const size_t pbase = ((size_t)gh * NCHUNK + chunk) * HD;
    for (int d = lane; d < HD; d += 32)
      part_acc[pbase + d] = v_new[kvh * HD + d];
    return;
  }

  // ---- cached 256-key chunk ----
  const int blk = block_idx[chunk];
  const size_t kv_base = ((size_t)blk * BLOCKK * NKV + kvh) * HD; // + key*NKV*HD

  // Phase 1: scores. Wave handles keys [wave*64, wave*64+64) as 4 WMMA tiles.
  const int off0 = (lane < 16) ? 0 : 2;    // A/B K-slot per half-wave (even -> 8B aligned)
  const int ncol = lane & 15;              // output column (q head if < 4)
  for (int t = 0; t < 4; ++t) {
    const int kb  = wave * 64 + t * 16;
    const int key = kb + (lane & 15);      // A-matrix row M = lane%16
    const float* __restrict__ krow = k_cache + kv_base + (size_t)key * NKV * HD;
    v8f c = {0.f, 0.f, 0.f, 0.f, 0.f, 0.f, 0.f, 0.f};
#pragma unroll 8
    for (int kk = 0; kk < HD / 4; ++kk) {
      v2f a = *(const v2f*)&krow[4 * kk + off0];        // global_load_b64
      v2f b = *(const v2f*)&q_s[ncol][4 * kk + off0];   // ds_load_b64, unconditional
#if __has_builtin(__builtin_amdgcn_wmma_f32_16x16x4_f32)
      c = __builtin_amdgcn_wmma_f32_16x16x4_f32(false, a, false, b,
                                                (short)0, c, false, false);
#else
      (void)a; (void)b;
#endif
    }
#if __has_builtin(__builtin_amdgcn_wmma_f32_16x16x4_f32)
    // D layout: lanes 0-15 hold rows M=r (VGPR r), lanes 16-31 rows M=8+r.
    if (ncol < NREP) {
      const int key0 = kb + ((lane < 16) ? 0 : 8);
#pragma unroll
      for (int r = 0; r < 8; ++r)
        s_s[ncol][key0 + r] = c[r] * SCALE;
    }
#else
    if (t == 0) { // scalar fallback (host pass / builtin absent)
      for (int idx = lane; idx < 64 * NREP; idx += 32) {
        const int key2 = wave * 64 + (idx >> 2);
        const int qh2  = idx & 3;
        const float* __restrict__ kr2 = k_cache + kv_base + (size_t)key2 * NKV * HD;
        float s = 0.f;
        for (int d = 0; d < HD; ++d) s = fmaf(kr2[d], q_s[qh2][d], s);
        s_s[qh2][key2] = s * SCALE;
      }
    }
#endif
  }
  __syncthreads();

  // Phase 2: per-chunk softmax partials + P*V. Wave == query head.
  const int qh = wave;
  float m = -1e30f;
  for (int key = lane; key < BLOCKK; key += 32)
    m = fmaxf(m, s_s[qh][key]);
#pragma unroll
  for (int off = 16; off > 0; off >>= 1)
    m = fmaxf(m, __shfl_xor(m, off, 32));

  float l = 0.f;
  for (int key = lane; key < BLOCKK; key += 32) {
    float p = __expf(s_s[qh][key] - m);
    s_s[qh][key] = p;
    l += p;
  }
#pragma unroll
  for (int off = 16; off > 0; off >>= 1)
    l += __shfl_xor(l, off, 32);

  // lane owns dims [lane*4, lane*4+4): coalesced b128 V reads
  float4 acc = make_float4(0.f, 0.f, 0.f, 0.f);
  for (int key = 0; key < BLOCKK; ++key) {
    const float p = s_s[qh][key];                 // LDS broadcast
    const float4* __restrict__ vrow =
        (const float4*)(v_cache + kv_base + (size_t)key * NKV * HD);
    float4 vv = vrow[lane];
    acc.x = fmaf(p, vv.x, acc.x);
    acc.y = fmaf(p, vv.y, acc.y);
    acc.z = fmaf(p, vv.z, acc.z);
    acc.w = fmaf(p, vv.w, acc.w);
  }

  const int gh = kvh * NREP + qh;
  const size_t pbase = ((size_t)gh * NCHUNK + chunk) * HD;
  ((float4*)(part_acc + pbase))[lane] = acc;
  if (lane == 0) {
    part_m[gh * NCHUNK + chunk] = m;
    part_l[gh * NCHUNK + chunk] = l;
  }
}

// ---------------------------------------------------------------------------
// Combine NCHUNK partials per query head (log-sum-exp merge).
// Grid: 32 blocks x 128 threads (thread = dim).
// ---------------------------------------------------------------------------
__global__ void attn_combine_kernel(const float* __restrict__ part_m,
                                    const float* __restrict__ part_l,
                                    const float* __restrict__ part_acc,
                                    float* __restrict__ attn_out) {
  const int gh = blockIdx.x;
  const int d  = threadIdx.x;
  float M = -1e30f;
#pragma unroll
  for (int c = 0; c < NCHUNK; ++c)
    M = fmaxf(M, part_m[gh * NCHUNK + c]);
  float L = 0.f, acc = 0.f;
#pragma unroll
  for (int c = 0; c < NCHUNK; ++c) {
    const float w = __expf(part_m[gh * NCHUNK + c] - M);
    L = fmaf(w, part_l[gh * NCHUNK + c], L);
    acc = fmaf(w, part_acc[((size_t)gh * NCHUNK + c) * HD + d], acc);
  }
  attn_out[gh * HD + d] = acc / L;
}

// ---------------------------------------------------------------------------
extern "C" void kernel_launch(void* const* d_in, const int* in_sizes, int n_in,
                              void* d_out, int out_size, void* d_ws, size_t ws_size,
                              hipStream_t stream) {
  const float* x    = (const float*)d_in[0];
  const float* fcos = (const float*)d_in[1];   // [64]
  const float* fsin = (const float*)d_in[2];   // [64]
  const float* wq   = (const float*)d_in[3];   // [4096][4096]
  const float* wk   = (const float*)d_in[4];   // [1024][4096]
  const float* wv   = (const float*)d_in[5];   // [1024][4096]
  const float* wo   = (const float*)d_in[6];   // [4096][4096]
  const float* kc   = (const float*)d_in[7];   // [128][256][8][128]
  const float* vc   = (const float*)d_in[8];
  const int*   bidx = (const int*)d_in[9];     // [16]

  float* ws   = (float*)d_ws;
  float* q    = ws;              // 4096
  float* k    = ws + 4096;       // 1024
  float* v    = ws + 5120;       // 1024
  float* pm   = ws + 6144;       // 32*17 = 544
  float* pl   = ws + 6688;       // 544
  float* pacc = ws + 7232;       // 32*17*128 = 69632
  float* aout = ws + 76864;      // 4096

  // 1) QKV projections (wave-per-row GEMV, 8 rows per 256-thread block)
  gemv4096_kernel<<<DIM / 8, 256, 0, stream>>>(wq, x, q, DIM);
  gemv4096_kernel<<<(NKV * HD) / 8, 256, 0, stream>>>(wk, x, k, NKV * HD);
  gemv4096_kernel<<<(NKV * HD) / 8, 256, 0, stream>>>(wv, x, v, NKV * HD);

  // 2) RoPE on q, k
  rope_kernel<<<10, 256, 0, stream>>>(q, k, fcos, fsin);

  // 3) Flash-decode partials over (kv head, key chunk)
  attn_partial_kernel<<<NKV * NCHUNK, 128, 0, stream>>>(
      q, k, v, kc, vc, bidx, pm, pl, pacc);

  // 4) Merge partials
  attn_combine_kernel<<<NH, HD, 0, stream>>>(pm, pl, pacc, aout);

  // 5) Output projection
  gemv4096_kernel<<<DIM / 8, 256, 0, stream>>>(wo, aout, (float*)d_out, DIM);
}